// NGCF_Encoder_12610023981255
// MI455X (gfx1250) — compile-verified
//
#include <hip/hip_runtime.h>

// NGCF encoder for MI455X (gfx1250).
// Memory plan: all feature matrices (38.4 MB each) live in d_ws and stay L2-resident
// (192 MB L2). SpMM = wave-per-edge, coalesced 128B row gathers + f32 L2 atomics.
// Dense 64-wide GEMMs use native fp32 WMMA (v_wmma_f32_16x16x4_f32); LDS tiles are
// staged by the Tensor Data Mover (tensor_load_to_lds, TENSORcnt) where possible,
// using TDM padding to produce the stride-68 bank-conflict-free A layout.

#define USER_NUM 100000
#define ITEM_NUM 50000
#define N_NODES  (USER_NUM + ITEM_NUM)
#define EMB      64
#define LAYERS   3
#define NNZ_EDGES 2400000
#define NE       (N_NODES * EMB)      // 9,600,000 floats per feature matrix

typedef float v2f __attribute__((ext_vector_type(2)));
typedef float v8f __attribute__((ext_vector_type(8)));

#if __has_builtin(__builtin_amdgcn_tensor_load_to_lds)
#define HAVE_TDM 1
typedef unsigned int u32x4 __attribute__((ext_vector_type(4)));
typedef int          i32x8 __attribute__((ext_vector_type(8)));
typedef int          i32x4 __attribute__((ext_vector_type(4)));

// Issue one 2D TDM load: global (row-major, 4-byte elems) -> LDS.
// OOB rows (>= tensor_d1) read as zero. Optional LDS padding:
// pad_interval_enc: 0=2,1=4,...,5=64,...,7=256 DWORDs between pads
// pad_amount_enc:   0=1,1=2,...  DWORDs inserted per pad
__device__ __forceinline__ void tdm_load_2d(const void* gaddr, unsigned lds_off,
                                            unsigned tensor_d0, unsigned tensor_d1,
                                            unsigned tile_d0, unsigned tile_d1,
                                            unsigned d0_stride,
                                            bool pad_en, unsigned pad_interval_enc,
                                            unsigned pad_amount_enc) {
    unsigned long long ga = (unsigned long long)gaddr;
    u32x4 g0;
    g0.x = 1u;                                       // count=1, user mode, no gather
    g0.y = lds_off;                                  // D#.lds_addr (bytes)
    g0.z = (unsigned)(ga & 0xffffffffu);             // global_addr[31:0]
    g0.w = (unsigned)((ga >> 32) & 0x01ffffffu)      // global_addr[56:32]
         | (2u << 30);                               // type=2 ("image")
    i32x8 g1;
    g1[0] = (int)((2u << 16)                         // data_size: 4 bytes
         | (pad_en ? (1u << 20) : 0u)                // pad_enable
         | (pad_interval_enc << 22)
         | (pad_amount_enc << 25));
    g1[1] = (int)(tensor_d0 << 16);                  // tensor_dim0[15:0]
    g1[2] = (int)((tensor_d0 >> 16) | (tensor_d1 << 16));
    g1[3] = (int)((tensor_d1 >> 16) | (tile_d0 << 16));
    g1[4] = (int)(tile_d1 & 0xffffu);                // tile_dim1 (tile_dim2=0)
    g1[5] = (int)d0_stride;                          // tensor_dim0_stride[31:0]
    g1[6] = 0;
    g1[7] = 0;
    i32x4 z4 = {0, 0, 0, 0};
    i32x8 z8 = {0, 0, 0, 0, 0, 0, 0, 0};
    // amdgpu-toolchain (clang-23 / therock-10.0) 6-arg form:
    //   (u32x4 g0, i32x8 g1, i32x4 g2, i32x4 g3, i32x8 extra, i32 cpol)
    __builtin_amdgcn_tensor_load_to_lds(g0, g1, z4, z4, z8, 0);
}
#else
#define HAVE_TDM 0
#endif

__device__ __forceinline__ unsigned lds_offset_of(const void* p) {
    // generic pointer to LDS: low 32 bits are the LDS byte offset
    return (unsigned)(unsigned long long)(size_t)p;
}

// ---------------------------------------------------------------------------
// ego = [user_emb ; item_emb];  sum(d_out) = ego
// ---------------------------------------------------------------------------
__global__ void concat_init(const float4* __restrict__ ue,
                            const float4* __restrict__ ie,
                            float4* __restrict__ ego,
                            float4* __restrict__ sum) {
    int i = blockIdx.x * blockDim.x + threadIdx.x;
    const int nu  = USER_NUM * EMB / 4;
    const int tot = NE / 4;
    if (i >= tot) return;
    float4 v = (i < nu) ? ue[i] : ie[i - nu];
    ego[i] = v;
    sum[i] = v;
}

// ---------------------------------------------------------------------------
// zero both aggregation buffers
// ---------------------------------------------------------------------------
__global__ void zero2(float4* __restrict__ a, float4* __restrict__ b) {
    int i = blockIdx.x * blockDim.x + threadIdx.x;
    if (i >= NE / 4) return;
    float4 z = make_float4(0.f, 0.f, 0.f, 0.f);
    a[i] = z;
    b[i] = z;
}

// ---------------------------------------------------------------------------
// Fused dual SpMM scatter: for each edge e,
//   aggT[rows[e],:] += vals[e] * temp[cols[e],:]
//   aggE[rows[e],:] += vals[e] * ego [cols[e],:]
// One wave32 per edge: lane handles dims {lane, lane+32} -> two coalesced
// 128B transactions per row per matrix. Scatter via global_atomic_add_f32;
// feature matrices are L2-resident so atomics stay on-chip.
// ---------------------------------------------------------------------------
__global__ __launch_bounds__(256)
void spmm_dual(const int*   __restrict__ rows,
               const int*   __restrict__ cols,
               const float* __restrict__ vals,
               const float* __restrict__ x1,   // temp
               const float* __restrict__ x2,   // ego
               float*       __restrict__ y1,   // aggT
               float*       __restrict__ y2) { // aggE
    int lane   = threadIdx.x & 31;
    int wave   = (blockIdx.x * blockDim.x + threadIdx.x) >> 5;
    int nwaves = (gridDim.x * blockDim.x) >> 5;
    for (int e = wave; e < NNZ_EDGES; e += nwaves) {
        int   r = rows[e];
        int   c = cols[e];
        float v = vals[e];
        int cb = c * EMB + lane;
        int rb = r * EMB + lane;
        float t0 = x1[cb], t1 = x1[cb + 32];
        float e0 = x2[cb], e1 = x2[cb + 32];
        atomicAdd(&y1[rb],      v * t0);
        atomicAdd(&y1[rb + 32], v * t1);
        atomicAdd(&y2[rb],      v * e0);
        atomicAdd(&y2[rb + 32], v * e1);
    }
}

// ---------------------------------------------------------------------------
// WMMA helper: 16x16 fp32 tile GEMM over K=64 from LDS (A stride 68 floats to
// avoid bank conflicts on the column-striped A-fragment reads).
// A fragment (16x4 f32, 2 VGPRs): lanes 0-15 hold K={j,j+1}, lanes 16-31 K={j+2,j+3}.
// ---------------------------------------------------------------------------
__device__ __forceinline__
v8f wmma_tile_k64(const float* __restrict__ sA, const float* __restrict__ sW,
                  int r0, int c0, int lane) {
    int half = lane >> 4;      // 0 or 1
    int l    = lane & 15;
    v8f acc = {};
#pragma unroll
    for (int j = 0; j < 64; j += 4) {
        v2f a, b;
        const float* ap = &sA[(r0 + l) * 68 + j + 2 * half];
        a.x = ap[0];
        a.y = ap[1];
        b.x = sW[(j + 2 * half) * 64 + c0 + l];
        b.y = sW[(j + 2 * half + 1) * 64 + c0 + l];
        acc = __builtin_amdgcn_wmma_f32_16x16x4_f32(
            /*neg_a=*/false, a, /*neg_b=*/false, b,
            /*c_mod=*/(short)0, acc, /*reuse_a=*/false, /*reuse_b=*/false);
    }
    return acc;
}

// ---------------------------------------------------------------------------
// temp = ego @ W   ([N,64] x [64,64]).  Block = 256 thr = 8 waves handling a
// 32-row x 64-col output macrotile (2 row-tiles x 4 col-tiles of 16x16).
// W and the A tile are staged into LDS by the Tensor Data Mover; TDM padding
// (64 DWORDs data + 4 DWORDs pad) produces the stride-68 A layout, and
// tensor_dim1 = N-rowBase makes OOB rows read as zero (replaces the guard).
// ---------------------------------------------------------------------------
__global__ __launch_bounds__(256)
void gemm_x_w(const float* __restrict__ X,
              const float* __restrict__ W,
              float* __restrict__ Y) {
    __shared__ float sA[32 * 68];
    __shared__ float sW[64 * 64];
    int tid = threadIdx.x;
    int rowBase = blockIdx.x * 32;

#if HAVE_TDM
    if (tid == 0) {
        tdm_load_2d(W, lds_offset_of(sW), 64, 64, 64, 64, 64,
                    /*pad=*/false, 0, 0);
        tdm_load_2d(X + (size_t)rowBase * EMB, lds_offset_of(sA),
                    64, (unsigned)(N_NODES - rowBase), 64, 32, 64,
                    /*pad=*/true, /*interval 64 dw=*/5, /*amount 4 dw=*/3);
        __builtin_amdgcn_s_wait_tensorcnt(0);
    }
    __syncthreads();
#else
    for (int i = tid; i < 512; i += 256) {          // 32x64 A tile, f4 chunks
        int r = i >> 4, c4 = i & 15;
        int g = rowBase + r;
        float4 v = make_float4(0.f, 0.f, 0.f, 0.f);
        if (g < N_NODES) v = ((const float4*)X)[g * 16 + c4];
        *(float4*)&sA[r * 68 + c4 * 4] = v;
    }
    for (int i = tid; i < 1024; i += 256)           // 64x64 W
        *(float4*)&sW[i * 4] = ((const float4*)W)[i];
    __syncthreads();
#endif

    int wave = tid >> 5, lane = tid & 31;
    int r0 = (wave >> 2) * 16;
    int c0 = (wave & 3) * 16;
    v8f acc = wmma_tile_k64(sA, sW, r0, c0, lane);

    int half = lane >> 4, l = lane & 15;
#pragma unroll
    for (int i = 0; i < 8; ++i) {
        int gr = rowBase + r0 + i + 8 * half;
        if (gr < N_NODES) Y[gr * 64 + c0 + l] = acc[i];
    }
}

// ---------------------------------------------------------------------------
// Fused layer tail:
//   P        = aggE (*) ego            (elementwise, built during LDS staging)
//   ego_new  = leaky_relu(aggT + temp + P @ W2)   (written in-place into ego)
//   sum     += ego_new
// W2 staged by TDM; P must be computed, so it is staged by the vector pipe
// (which overlaps with the in-flight TDM). In-place ego write is safe: the
// block stages its own 32 ego rows into LDS before any write.
// ---------------------------------------------------------------------------
__global__ __launch_bounds__(256)
void layer_tail(float*       __restrict__ ego,
                const float* __restrict__ aggE,
                const float* __restrict__ aggT,
                const float* __restrict__ temp,
                const float* __restrict__ W2,
                float*       __restrict__ sum) {
    __shared__ float sA[32 * 68];
    __shared__ float sW[64 * 64];
    int tid = threadIdx.x;
    int rowBase = blockIdx.x * 32;

#if HAVE_TDM
    if (tid == 0)
        tdm_load_2d(W2, lds_offset_of(sW), 64, 64, 64, 64, 64,
                    /*pad=*/false, 0, 0);
#endif
    for (int i = tid; i < 512; i += 256) {
        int r = i >> 4, c4 = i & 15;
        int g = rowBase + r;
        float4 v = make_float4(0.f, 0.f, 0.f, 0.f);
        if (g < N_NODES) {
            float4 e  = ((const float4*)ego )[g * 16 + c4];
            float4 ae = ((const float4*)aggE)[g * 16 + c4];
            v = make_float4(e.x * ae.x, e.y * ae.y, e.z * ae.z, e.w * ae.w);
        }
        *(float4*)&sA[r * 68 + c4 * 4] = v;
    }
#if HAVE_TDM
    if (tid == 0) __builtin_amdgcn_s_wait_tensorcnt(0);
#else
    for (int i = tid; i < 1024; i += 256)
        *(float4*)&sW[i * 4] = ((const float4*)W2)[i];
#endif
    __syncthreads();

    int wave = tid >> 5, lane = tid & 31;
    int r0 = (wave >> 2) * 16;
    int c0 = (wave & 3) * 16;
    v8f acc = wmma_tile_k64(sA, sW, r0, c0, lane);

    int half = lane >> 4, l = lane & 15;
#pragma unroll
    for (int i = 0; i < 8; ++i) {
        int gr = rowBase + r0 + i + 8 * half;
        if (gr < N_NODES) {
            int idx = gr * 64 + c0 + l;
            float o = acc[i] + aggT[idx] + temp[idx];
            o = (o > 0.f) ? o : 0.01f * o;          // leaky_relu, slope 0.01
            ego[idx]  = o;
            sum[idx] += o;
        }
    }
}

// ---------------------------------------------------------------------------
// mean over (LAYERS+1)=4 stacked embeddings
// ---------------------------------------------------------------------------
__global__ void finalize(float4* __restrict__ out) {
    int i = blockIdx.x * blockDim.x + threadIdx.x;
    if (i >= NE / 4) return;
    float4 v = out[i];
    out[i] = make_float4(v.x * 0.25f, v.y * 0.25f, v.z * 0.25f, v.w * 0.25f);
}

// ---------------------------------------------------------------------------
extern "C" void kernel_launch(void* const* d_in, const int* in_sizes, int n_in,
                              void* d_out, int out_size, void* d_ws, size_t ws_size,
                              hipStream_t stream) {
    (void)in_sizes; (void)n_in; (void)out_size; (void)ws_size;

    const float* user_emb = (const float*)d_in[0];
    const float* item_emb = (const float*)d_in[1];
    const float* w1       = (const float*)d_in[2];   // [3,64,64]
    const float* w2       = (const float*)d_in[3];   // [3,64,64]
    const float* adj_vals = (const float*)d_in[4];
    const int*   adj_rows = (const int*)d_in[5];
    const int*   adj_cols = (const int*)d_in[6];
    float* out = (float*)d_out;                      // doubles as running sum

    // Workspace: 4 feature matrices, 38.4 MB each (153.6 MB total)
    float* ego  = (float*)d_ws;
    float* temp = ego  + NE;
    float* aggT = temp + NE;
    float* aggE = aggT + NE;

    const int ELEM_BLOCKS = (NE / 4 + 255) / 256;
    const int GEMM_BLOCKS = (N_NODES + 31) / 32;

    concat_init<<<ELEM_BLOCKS, 256, 0, stream>>>(
        (const float4*)user_emb, (const float4*)item_emb,
        (float4*)ego, (float4*)out);

    for (int k = 0; k < LAYERS; ++k) {
        const float* w1k = w1 + k * EMB * EMB;
        const float* w2k = w2 + k * EMB * EMB;

        gemm_x_w<<<GEMM_BLOCKS, 256, 0, stream>>>(ego, w1k, temp);
        zero2<<<ELEM_BLOCKS, 256, 0, stream>>>((float4*)aggT, (float4*)aggE);
        spmm_dual<<<2048, 256, 0, stream>>>(adj_rows, adj_cols, adj_vals,
                                            temp, ego, aggT, aggE);
        layer_tail<<<GEMM_BLOCKS, 256, 0, stream>>>(ego, aggE, aggT, temp,
                                                    w2k, out);
    }

    finalize<<<ELEM_BLOCKS, 256, 0, stream>>>((float4*)out);
}